// MockTopKRouter_6562710028727
// MI455X (gfx1250) — compile-verified
//
#include <hip/hip_runtime.h>
#include <hip/hip_bf16.h>

typedef __attribute__((ext_vector_type(2))) float v2f;
typedef __attribute__((ext_vector_type(8))) float v8f;

#define HIDDEN   4096
#define NEXPERTS 64
#define KC       256      // K-chunk staged in LDS
#define BPITCH   260      // floats; 260 mod 64 = 4 -> conflict-free b64 B reads
#define LPITCH   65       // floats; conflict-free per-row logit reads
#define ROWS_PER_WG 128   // 8 waves x 16 rows

__global__ __launch_bounds__(256)
void moe_router_kernel(const float* __restrict__ x,
                       const float* __restrict__ gate_w,
                       const float* __restrict__ gate_b,
                       float* __restrict__ rweights,
                       int*   __restrict__ sel,
                       float* __restrict__ logits,
                       float* __restrict__ aux)
{
    __shared__ float smem[NEXPERTS * BPITCH];   // 66,560 B; reused for logits tile

    const int tid  = threadIdx.x;
    const int wave = tid >> 5;          // 0..7 (wave32)
    const int lane = tid & 31;
    const int lrow = lane & 15;         // M (A) / N (B,C) within tile
    const int lhi  = lane >> 4;         // half-wave select

    const int rowBase = blockIdx.x * ROWS_PER_WG + wave * 16;
    // A-fragment pointer: lane holds x[rowBase+lrow][k + 2*lhi .. +1]
    const float* aptr = x + (size_t)(rowBase + lrow) * HIDDEN + 2 * lhi;

    const float4* gw4 = reinterpret_cast<const float4*>(gate_w);

    v8f acc[4] = {};

    for (int k0 = 0; k0 < HIDDEN; k0 += KC) {
        // ---- stage W[:, k0:k0+KC] into LDS (row-major, padded pitch) ----
        for (int i = tid; i < NEXPERTS * (KC / 4); i += 256) {
            const int e  = i >> 6;        // KC/4 == 64 float4 per row
            const int c4 = i & 63;
            float4 v = gw4[(size_t)e * (HIDDEN / 4) + (k0 >> 2) + c4];
            *reinterpret_cast<float4*>(&smem[e * BPITCH + (c4 << 2)]) = v;
        }
        __syncthreads();

        if (k0 + KC < HIDDEN)
            __builtin_prefetch(aptr + k0 + KC, 0, 0);   // global_prefetch_b8

        // ---- fp32 WMMA: 16x16x4, 4 N-tiles share one A fragment ----
        #pragma unroll 8
        for (int kk = 0; kk < KC; kk += 4) {
            v2f a = *reinterpret_cast<const v2f*>(aptr + k0 + kk);
            #pragma unroll
            for (int t = 0; t < 4; ++t) {
                v2f b = *reinterpret_cast<const v2f*>(
                    &smem[(t * 16 + lrow) * BPITCH + kk + 2 * lhi]);
                acc[t] = __builtin_amdgcn_wmma_f32_16x16x4_f32(
                    false, a, false, b, (short)0, acc[t], false, false);
            }
        }
        __syncthreads();
    }

    // ---- epilogue: bias, emit logits, stage tile for top-2 ----
    float bias[4];
    #pragma unroll
    for (int t = 0; t < 4; ++t) bias[t] = gate_b[t * 16 + lrow];

    const size_t gRow0 = (size_t)blockIdx.x * ROWS_PER_WG;
    #pragma unroll
    for (int t = 0; t < 4; ++t) {
        const int e = t * 16 + lrow;                 // expert (C-matrix N)
        #pragma unroll
        for (int r = 0; r < 8; ++r) {
            const int lr = wave * 16 + 8 * lhi + r;  // local row (C-matrix M)
            const float v = acc[t][r] + bias[t];
            logits[(gRow0 + lr) * NEXPERTS + e] = v;
            smem[lr * LPITCH + e] = v;
        }
    }
    __syncthreads();

    if (tid < ROWS_PER_WG) {
        const float* row = &smem[tid * LPITCH];
        float b1 = -3.402823466e38f, b2 = -3.402823466e38f;
        int   i1 = 0, i2 = 0;
        #pragma unroll 8
        for (int e = 0; e < NEXPERTS; ++e) {
            const float v = row[e];
            if (v > b1)      { b2 = b1; i2 = i1; b1 = v; i1 = e; }
            else if (v > b2) { b2 = v; i2 = e; }
        }
        const float e1 = expf(b2 - b1);       // softmax over [b1, b2], b1 max
        const float w0 = 1.0f / (1.0f + e1);
        const size_t g = gRow0 + tid;
        rweights[g * 2 + 0] = w0;
        rweights[g * 2 + 1] = e1 * w0;
        sel[g * 2 + 0] = i1;
        sel[g * 2 + 1] = i2;
    }

    if (blockIdx.x == 0 && tid == 0) aux[0] = 0.0f;
}

extern "C" void kernel_launch(void* const* d_in, const int* in_sizes, int n_in,
                              void* d_out, int out_size, void* d_ws, size_t ws_size,
                              hipStream_t stream) {
    const float* x  = (const float*)d_in[0];
    const float* gw = (const float*)d_in[1];
    const float* gb = (const float*)d_in[2];

    const int N = in_sizes[0] / HIDDEN;      // 16384 tokens

    float* out     = (float*)d_out;
    float* rw      = out;                                // [N,2] f32
    int*   sel     = (int*)(out + (size_t)N * 2);        // [N,2] i32
    float* logits  = out + (size_t)N * 4;                // [N,64] f32
    float* aux     = out + (size_t)N * 68;               // scalar

    const int grid = N / ROWS_PER_WG;        // 128 workgroups
    moe_router_kernel<<<grid, 256, 0, stream>>>(x, gw, gb, rw, sel, logits, aux);
}